// model_8830452760818
// MI455X (gfx1250) — compile-verified
//
#include <hip/hip_runtime.h>

// Per-particle SDE rollout with boundary-exit quartic (Cardano) correction.
// VALU-bound streaming kernel: the Cardano solve is gated behind a per-lane
// mask (execz-skipped when no lane in the wave exits that step); the whole
// wave breaks out of the time loop once every lane is frozen. dw loads are
// software-pipelined one step ahead. Memory traffic = the 206MB input floor.
// All math is call-free: cos -> v_cos_f32 (arg bounded by pi/6), cube root ->
// v_log/v_exp (matches XLA's pow(x,1/3) lowering), exp -> v_exp_f32.

#define QEPS 1e-12f

__device__ __forceinline__ float sgnf(float v) {
    return (v > 0.0f) ? 1.0f : ((v < 0.0f) ? -1.0f : 0.0f);
}

// One step's boundary-exit correction. drift=(d0,d1), diff=(f0,f1), state=(x0v,x1v).
// Cold path keeps the reference's exact IEEE-division structure for fidelity —
// it only runs for lanes exiting this step, so correctness > throughput here.
__device__ __forceinline__ void quartic_coef(
    float x0v, float x1v, float d0, float d1, float f0, float f1,
    float flag, float& coef, float& Exit)
{
    float xt0 = x0v + d0 + f0;
    float xt1 = x1v + d1 + f1;
    float bval = xt0 * xt0 + xt1 * xt1 - 1.0f;      // R^2 = 1
    Exit = (bval >= 0.0f) ? 1.0f : 0.0f;            // ceil((sign+1)/2)
    coef = (1.0f - Exit) * flag;
    bool mask = (flag * Exit) == 1.0f;              // lanes exiting THIS step

    if (__builtin_expect(mask, 0)) {                // execz-skipped when none exit
        float a = d0 * d0 + d1 * d1;
        float b = 2.0f * (d0 * f0 + d1 * f1);
        float c = 2.0f * (d0 * x0v + d1 * x1v) + f0 * f0 + f1 * f1;
        float d = 2.0f * (f0 * x0v + f1 * x1v);
        float e = x0v * x0v + x1v * x1v - 1.0f;

        float a_s = fmaxf(a, QEPS);
        float p = (8.0f * a * c - 3.0f * b * b) / (8.0f * a_s * a_s);
        float q = (b * b * b - 4.0f * a * b * c + 8.0f * a * a * d)
                / (8.0f * a_s * a_s * a_s);
        float sign_q = sgnf(q);

        float D0 = c * c - 3.0f * b * d + 12.0f * a * e;
        float D1 = 2.0f * c * c * c - 9.0f * b * c * d + 27.0f * b * b * e
                 + 27.0f * a * d * d - 72.0f * a * c * e;
        float D2 = D1 * D1 - 4.0f * D0 * D0 * D0;
        bool plusBranch = (D2 >= 0.0f);             // signal_D2

        float QQ = 0.5f * (D1 + sqrtf(fabsf(D2)));
        // sign(QQ)*|QQ|^(1/3): exp2(log2(|QQ|)/3) — call-free, matches XLA pow
        // lowering; |QQ|==0 -> log2=-inf -> exp2=0, correct limit.
        float Q  = sgnf(QQ) * exp2f(log2f(fabsf(QQ)) * 0.33333334f);
        float Q_s = (fabsf(Q) > QEPS) ? Q : QEPS;

        float two_p_3 = 2.0f * p / 3.0f;
        float S_plus = 0.5f * sqrtf(fabsf((Q + D0 / Q_s) / (3.0f * a_s) - two_p_3));

        float D0_s  = (fabsf(D0) > QEPS) ? D0 : QEPS;
        float ratio = (D1 * D1) / (4.0f * D0_s * D0_s * D0_s);
        float phi   = acosf(fminf(sqrtf(fabsf(ratio)), 1.0f));
        // phi in [0, pi/2] -> phi/3 in [0, pi/6]: no range reduction needed,
        // hardware v_cos_f32 is exact enough and avoids the __ocml_cos call.
        float S_minus = 0.5f * sqrtf(fabsf(
            2.0f * sqrtf(fabsf(D0)) * __cosf(phi / 3.0f) / (3.0f * a_s) - two_p_3));

        float S   = plusBranch ? S_plus : S_minus;
        float S_s = (fabsf(S) > QEPS) ? S : QEPS;
        float aqs = fabsf(q / S_s);

        float base = -4.0f * S * S - 2.0f * p;
        float boa  = b / (4.0f * a_s);
        float sqrt_rho = 0.5f * sqrtf(fabsf(base + aqs)) - boa - sign_q * S;
        bool  bad = ((1.0f - sqrt_rho) * sqrt_rho) < 0.0f;
        float new_sqrt_rho = 0.5f * sqrtf(fabsf(base - aqs)) - boa + sign_q * S;
        float srf = bad ? new_sqrt_rho : sqrt_rho;
        coef = srf * srf;                            // rho
    }
}

__global__ __launch_bounds__(256) void sde_exit_kernel(
    const float* __restrict__ x0g,   // [B,2]
    const float* __restrict__ dwg,   // [B,2,N]
    const float* __restrict__ ug,    // scalar
    float* __restrict__ out,         // [B]
    int Btot, int Nsteps)
{
    const int b = blockIdx.x * blockDim.x + threadIdx.x;
    if (b >= Btot) return;

    const float u     = *ug;
    const float dt    = 0.2f / (float)Nsteps;        // T / N
    const float sigma = 1.41421356237309515f;        // sqrt(2)
    const float wc    = u * u + 2.0f;                // (u^2 + 2)

    const float* dwp = dwg + (size_t)b * (size_t)(2 * Nsteps);
    // gfx1250 global_prefetch_b8: pull this thread's 400B noise stream early
    __builtin_prefetch(dwp + 0,  0, 3);
    __builtin_prefetch(dwp + 32, 0, 3);
    __builtin_prefetch(dwp + 64, 0, 3);
    __builtin_prefetch(dwp + 96, 0, 3);

    float x0c = x0g[2 * b + 0];
    float x1c = x0g[2 * b + 1];

    // ---- step 0: coef computed, but x/flag NOT updated (reference quirk) ----
    float coef, Exit;
    {
        float d0 = u * x0c * dt, d1 = u * x1c * dt;
        float f0 = sigma * dwp[0], f1 = sigma * dwp[Nsteps];
        quartic_coef(x0c, x1c, d0, d1, f0, f1, 1.0f, coef, Exit);
    }
    float sq   = x0c * x0c + x1c * x1c;
    float w    = wc * sq - 4.0f;                     // -2*DIM
    float y    = coef * w * dt;                      // disc = exp(0) = 1
    float csum = coef;                               // inclusive prefix so far
    float flag = 1.0f;

    // ---- steps 1..N-1, streaming reduction, loads pipelined one step ahead ----
    float r0 = dwp[1];
    float r1 = dwp[Nsteps + 1];
    for (int i = 1; i < Nsteps; ++i) {
        float f0 = sigma * r0, f1 = sigma * r1;

        // issue next iteration's loads before the heavy math (clamped in-bounds)
        int nx = (i + 1 < Nsteps) ? (i + 1) : i;
        r0 = dwp[nx];
        r1 = dwp[Nsteps + nx];

        float d0 = u * x0c * dt, d1 = u * x1c * dt;
        quartic_coef(x0c, x1c, d0, d1, f0, f1, flag, coef, Exit);

        float sc = (coef > 0.0f) ? sqrtf(coef) : 0.0f;   // _safe_sqrt
        x0c = x0c + d0 * coef + f0 * sc;
        x1c = x1c + d1 * coef + f1 * sc;
        flag = flag * (1.0f - Exit);

        sq = x0c * x0c + x1c * x1c;
        w  = wc * sq - 4.0f;
        float disc = __expf(-dt * csum);             // gamma=1; exclusive prefix
        y += coef * w * dt * disc;
        csum += coef;

        // whole wave frozen -> every remaining term is exactly 0 / state frozen
        if (__ballot(flag != 0.0f) == 0ull) break;
    }

    y += __expf(-dt * csum) * sq;                    // terminal term
    out[b] = y;
}

extern "C" void kernel_launch(void* const* d_in, const int* in_sizes, int n_in,
                              void* d_out, int out_size, void* d_ws, size_t ws_size,
                              hipStream_t stream) {
    const float* x0 = (const float*)d_in[0];   // [B,2] f32
    const float* dw = (const float*)d_in[1];   // [B,2,N] f32
    const float* u  = (const float*)d_in[2];   // scalar f32

    float* out = (float*)d_out;                // [B,1] f32

    const int Btot   = in_sizes[0] / 2;
    const int Nsteps = (Btot > 0) ? in_sizes[1] / (2 * Btot) : 50;

    const int threads = 256;                   // 8 wave32 per block
    const int blocks  = (Btot + threads - 1) / threads;
    sde_exit_kernel<<<blocks, threads, 0, stream>>>(x0, dw, u, out, Btot, Nsteps);
}